// GNNBlock_2611340116282
// MI455X (gfx1250) — compile-verified
//
#include <hip/hip_runtime.h>
#include <math.h>

// ---------------- problem constants (match reference) ----------------
#define N_NODES  50000
#define DIM      128
#define HEADS    4
#define HEAD_DIM 32
#define N_EDGES  800000
#define LN_EPS   1e-5f

// LDS A-tile row padding: 128+4 floats = 528 B/row -> 16B aligned rows and
// per-row bank offset of 4 (64 banks) => conflict-free ds_load_b64.
#define A_STRIDE (DIM + 4)

typedef __attribute__((ext_vector_type(2))) float v2f;
typedef __attribute__((ext_vector_type(8))) float v8f;

// ---------------- helpers ----------------
// Order-preserving map float -> uint so atomicMax works for signed floats.
__device__ __forceinline__ unsigned int f2key(float f) {
    unsigned int b = __float_as_uint(f);
    return (b & 0x80000000u) ? ~b : (b | 0x80000000u);
}
__device__ __forceinline__ float key2f(unsigned int k) {
    unsigned int b = (k & 0x80000000u) ? (k & 0x7FFFFFFFu) : ~k;
    return __uint_as_float(b);
}

// ---------------- kernel 0: zero agg + reduction cells ----------------
__global__ __launch_bounds__(256)
void gnn_zero_init(float* __restrict__ agg, unsigned int* __restrict__ maxkey,
                   float* __restrict__ sums) {
    size_t i = (size_t)blockIdx.x * blockDim.x + threadIdx.x;
    const size_t total = (size_t)N_NODES * DIM;
    for (; i < total; i += (size_t)gridDim.x * blockDim.x) agg[i] = 0.0f;
    if (blockIdx.x == 0 && threadIdx.x < HEADS) {
        maxkey[threadIdx.x] = 0u;   // key 0 == smallest possible
        sums[threadIdx.x]   = 0.0f;
    }
}

// ---------------- kernel 1: C[nrows x 128] = A @ W + bias via WMMA f32 ----
// Block (8 waves) covers one 16-row tile x all 8 column tiles. The shared
// 16x128 A tile is staged to LDS with the gfx1250 async Global->LDS path
// (ASYNCcnt), then each wave runs 32 x V_WMMA_F32_16X16X4_F32 fed from LDS.
__global__ __launch_bounds__(256)
void gnn_gemm_wmma_f32(const float* __restrict__ A, const float* __restrict__ W,
                       const float* __restrict__ bias, float* __restrict__ C,
                       int nrows) {
    __shared__ float sA[16 * A_STRIDE];

    const int lane = threadIdx.x & 31;
    const int wave = threadIdx.x >> 5;       // = column tile 0..7
    const int r0 = blockIdx.x * 16;
    if (r0 >= nrows) return;                 // uniform (grid sized exactly)

    // ---- cooperative async copy: 16 rows x 512 B = 512 x 16 B chunks ----
    #pragma unroll
    for (int cc = 0; cc < 2; ++cc) {
        const int c    = threadIdx.x + cc * 256;   // chunk id 0..511
        const int row  = c >> 5;                   // 32 chunks per row
        const int colf = (c & 31) << 2;            // float column (x4)
        const unsigned lds_off =
            (unsigned)(size_t)(&sA[row * A_STRIDE + colf]);
        const unsigned long long gaddr =
            (unsigned long long)(A + (size_t)(r0 + row) * DIM + colf);
        asm volatile("global_load_async_to_lds_b128 %0, %1, off"
                     :: "v"(lds_off), "v"(gaddr) : "memory");
    }
    asm volatile("s_wait_asynccnt 0x0" ::: "memory");
    __syncthreads();

    // ---- WMMA accumulation over K = 128 in steps of 4 ----
    const int n0   = wave * 16;
    const int m    = lane & 15;              // M row (A) / N col (B)
    const int half = lane >> 4;              // k sub-pair select
    const float* __restrict__ srow = &sA[m * A_STRIDE];

    v8f acc = {};
    #pragma unroll 8
    for (int kb = 0; kb < DIM; kb += 4) {
        // A 16x4: lanes 0-15 -> v0=K0,v1=K1 ; lanes 16-31 -> v0=K2,v1=K3
        v2f a, b;
        a.x = srow[kb + 2 * half + 0];       // ds_load (conflict-free, padded)
        a.y = srow[kb + 2 * half + 1];
        // B 4x16 (row-major W[k][n]): lanes 0-15 -> rows K0/K1, 16-31 -> K2/K3
        b.x = W[(size_t)(kb + 2 * half + 0) * DIM + n0 + m];
        b.y = W[(size_t)(kb + 2 * half + 1) * DIM + n0 + m];
        acc = __builtin_amdgcn_wmma_f32_16x16x4_f32(
            /*neg_a=*/false, a, /*neg_b=*/false, b,
            /*c_mod=*/(short)0, acc, /*reuse_a=*/false, /*reuse_b=*/false);
    }

    // C/D layout: vgpr j, lanes 0-15 -> M=j ; lanes 16-31 -> M=j+8 ; N=lane&15
    const int col = n0 + m;
    const float bv = bias[col];
    #pragma unroll
    for (int j = 0; j < 8; ++j) {
        const int row = r0 + j + 8 * half;
        C[(size_t)row * DIM + col] = acc[j] + bv;
    }
}

// ---------------- kernel 2: per-edge head scores + global max ------------
__global__ __launch_bounds__(256)
void gnn_edge_scores(const float* __restrict__ Q, const float* __restrict__ K,
                     const int* __restrict__ ei, float* __restrict__ scores,
                     unsigned int* __restrict__ maxkey) {
    __shared__ unsigned int smax[HEADS];
    if (threadIdx.x < HEADS) smax[threadIdx.x] = 0u;
    __syncthreads();

    const int lane = threadIdx.x & 31;
    const int wave = threadIdx.x >> 5;
    const int e = blockIdx.x * 8 + wave;
    if (e < N_EDGES) {
        const int src = ei[e];
        const int dst = ei[N_EDGES + e];
        const float* __restrict__ q = Q + (size_t)src * DIM;
        const float* __restrict__ k = K + (size_t)dst * DIM;
        // lane covers dim lane+32h for head h -> 4 wave-wide dot products
        float p0 = q[lane     ] * k[lane     ];
        float p1 = q[lane + 32] * k[lane + 32];
        float p2 = q[lane + 64] * k[lane + 64];
        float p3 = q[lane + 96] * k[lane + 96];
        #pragma unroll
        for (int off = 16; off > 0; off >>= 1) {
            p0 += __shfl_xor(p0, off, 32);
            p1 += __shfl_xor(p1, off, 32);
            p2 += __shfl_xor(p2, off, 32);
            p3 += __shfl_xor(p3, off, 32);
        }
        if (lane == 0) {
            const float scale = 0.17677669529663687f;   // 1/sqrt(32)
            float s[HEADS] = {p0 * scale, p1 * scale, p2 * scale, p3 * scale};
            #pragma unroll
            for (int h = 0; h < HEADS; ++h) {
                float v = (s[h] >= 0.0f) ? s[h] : 0.2f * s[h];  // LeakyReLU(0.2)
                scores[(size_t)e * HEADS + h] = v;
                atomicMax(&smax[h], f2key(v));
            }
        }
    }
    __syncthreads();
    if (threadIdx.x < HEADS) atomicMax(&maxkey[threadIdx.x], smax[threadIdx.x]);
}

// ---------------- kernel 3: per-head sum of exp(score - max) -------------
__global__ __launch_bounds__(256)
void gnn_softmax_sum(const float* __restrict__ scores,
                     const unsigned int* __restrict__ maxkey,
                     float* __restrict__ sums) {
    __shared__ float ssum[HEADS];
    if (threadIdx.x < HEADS) ssum[threadIdx.x] = 0.0f;
    __syncthreads();

    float mx[HEADS];
    #pragma unroll
    for (int h = 0; h < HEADS; ++h) mx[h] = key2f(maxkey[h]);

    float acc[HEADS] = {0.f, 0.f, 0.f, 0.f};
    const size_t total = (size_t)N_EDGES * HEADS;
    size_t i = (size_t)blockIdx.x * blockDim.x + threadIdx.x;
    // stride is a multiple of 4 -> each thread keeps a fixed head slot
    for (; i < total; i += (size_t)gridDim.x * blockDim.x) {
        const int h = (int)(i & 3u);
        acc[h] += expf(scores[i] - mx[h]);
    }
    #pragma unroll
    for (int h = 0; h < HEADS; ++h)
        if (acc[h] != 0.0f) atomicAdd(&ssum[h], acc[h]);
    __syncthreads();
    if (threadIdx.x < HEADS) atomicAdd(&sums[threadIdx.x], ssum[threadIdx.x]);
}

// ---------------- kernel 4: weighted V[src] scatter-add into agg[dst] ----
__global__ __launch_bounds__(256)
void gnn_scatter_messages(const float* __restrict__ V, const int* __restrict__ ei,
                          const float* __restrict__ scores,
                          const unsigned int* __restrict__ maxkey,
                          const float* __restrict__ sums,
                          float* __restrict__ agg) {
    const int lane = threadIdx.x & 31;
    const int wave = threadIdx.x >> 5;
    const int e = blockIdx.x * 8 + wave;
    if (e >= N_EDGES) return;
    const int src = ei[e];
    const int dst = ei[N_EDGES + e];

    float w[HEADS];
    #pragma unroll
    for (int h = 0; h < HEADS; ++h) {
        const float mx = key2f(maxkey[h]);
        w[h] = expf(scores[(size_t)e * HEADS + h] - mx) / sums[h];
    }

    const float* __restrict__ v = V + (size_t)src * DIM;
    float* __restrict__ a = agg + (size_t)dst * DIM;
    unsafeAtomicAdd(&a[lane     ], w[0] * v[lane     ]);
    unsafeAtomicAdd(&a[lane + 32], w[1] * v[lane + 32]);
    unsafeAtomicAdd(&a[lane + 64], w[2] * v[lane + 64]);
    unsafeAtomicAdd(&a[lane + 96], w[3] * v[lane + 96]);
}

// ---------------- kernel 5: residual + LayerNorm (wave per row) ----------
__global__ __launch_bounds__(256)
void gnn_residual_ln(const float* __restrict__ x, const float* __restrict__ h,
                     const float* __restrict__ gamma, const float* __restrict__ beta,
                     float* __restrict__ out) {
    const int lane = threadIdx.x & 31;
    const int wave = threadIdx.x >> 5;
    const int row = blockIdx.x * 8 + wave;
    if (row >= N_NODES) return;
    const float* __restrict__ xr = x + (size_t)row * DIM;
    const float* __restrict__ hr = h + (size_t)row * DIM;

    float y0 = xr[lane     ] + hr[lane     ];
    float y1 = xr[lane + 32] + hr[lane + 32];
    float y2 = xr[lane + 64] + hr[lane + 64];
    float y3 = xr[lane + 96] + hr[lane + 96];

    float s = y0 + y1 + y2 + y3;
    #pragma unroll
    for (int off = 16; off > 0; off >>= 1) s += __shfl_xor(s, off, 32);
    const float mu = s * (1.0f / DIM);

    const float d0 = y0 - mu, d1 = y1 - mu, d2 = y2 - mu, d3 = y3 - mu;
    float var = d0 * d0 + d1 * d1 + d2 * d2 + d3 * d3;
    #pragma unroll
    for (int off = 16; off > 0; off >>= 1) var += __shfl_xor(var, off, 32);
    const float inv = rsqrtf(var * (1.0f / DIM) + LN_EPS);

    float* __restrict__ o = out + (size_t)row * DIM;
    o[lane     ] = d0 * inv * gamma[lane     ] + beta[lane     ];
    o[lane + 32] = d1 * inv * gamma[lane + 32] + beta[lane + 32];
    o[lane + 64] = d2 * inv * gamma[lane + 64] + beta[lane + 64];
    o[lane + 96] = d3 * inv * gamma[lane + 96] + beta[lane + 96];
}

// ---------------- host launcher ----------------
extern "C" void kernel_launch(void* const* d_in, const int* in_sizes, int n_in,
                              void* d_out, int out_size, void* d_ws, size_t ws_size,
                              hipStream_t stream) {
    (void)in_sizes; (void)n_in; (void)out_size; (void)ws_size;

    const float* x     = (const float*)d_in[0];
    const int*   ei    = (const int*)  d_in[1];   // [2, E] (row0=src, row1=dst)
    const float* Wq    = (const float*)d_in[2];
    const float* bq    = (const float*)d_in[3];
    const float* Wk    = (const float*)d_in[4];
    const float* bk    = (const float*)d_in[5];
    const float* Wv    = (const float*)d_in[6];
    const float* bv    = (const float*)d_in[7];
    const float* Wo    = (const float*)d_in[8];
    const float* bo    = (const float*)d_in[9];
    const float* gamma = (const float*)d_in[10];
    const float* beta  = (const float*)d_in[11];
    float* out = (float*)d_out;

    // workspace layout (floats)
    const size_t ND = (size_t)N_NODES * DIM;       // 6,400,000
    float* Qb     = (float*)d_ws;
    float* Kb     = Qb + ND;
    float* Vb     = Kb + ND;
    float* agg    = Vb + ND;
    float* scores = agg + ND;                      // E*HEADS = 3,200,000
    unsigned int* maxkey = (unsigned int*)(scores + (size_t)N_EDGES * HEADS);
    float* sums   = (float*)(maxkey + HEADS);
    float* hbuf   = Qb;                            // reuse Q after edge phase

    const int gemmBlocks = N_NODES / 16;                     // 3125 row tiles
    const int edgeBlocks = N_EDGES / 8;                      // 100000
    const int rowBlocks  = N_NODES / 8;                      // 6250

    gnn_zero_init<<<2048, 256, 0, stream>>>(agg, maxkey, sums);

    gnn_gemm_wmma_f32<<<gemmBlocks, 256, 0, stream>>>(x, Wq, bq, Qb, N_NODES);
    gnn_gemm_wmma_f32<<<gemmBlocks, 256, 0, stream>>>(x, Wk, bk, Kb, N_NODES);
    gnn_gemm_wmma_f32<<<gemmBlocks, 256, 0, stream>>>(x, Wv, bv, Vb, N_NODES);

    gnn_edge_scores<<<edgeBlocks, 256, 0, stream>>>(Qb, Kb, ei, scores, maxkey);
    gnn_softmax_sum<<<1024, 256, 0, stream>>>(scores, maxkey, sums);
    gnn_scatter_messages<<<edgeBlocks, 256, 0, stream>>>(Vb, ei, scores, maxkey, sums, agg);

    gnn_gemm_wmma_f32<<<gemmBlocks, 256, 0, stream>>>(agg, Wo, bo, hbuf, N_NODES);

    gnn_residual_ln<<<rowBlocks, 256, 0, stream>>>(x, hbuf, gamma, beta, out);
}